// IASA_34806414966812
// MI455X (gfx1250) — compile-verified
//
#include <hip/hip_runtime.h>
#include <hip/hip_bf16.h>

// Fixed problem shapes from the reference setup.
#define BATCH 2
#define SEQ   16384
#define DIM   256
#define QKD   256
#define NH    8
#define DQ    32
#define DV    32
#define GSZ   128
#define NGRP  128
#define MG    64

typedef __attribute__((ext_vector_type(2))) float v2f;
typedef __attribute__((ext_vector_type(8))) float v8f;

__device__ __forceinline__ v8f wmma4(v2f a, v2f b, v8f c) {
  // V_WMMA_F32_16X16X4_F32 : full fp32 MACs, D = A(16x4) * B(4x16) + C
  return __builtin_amdgcn_wmma_f32_16x16x4_f32(false, a, false, b, (short)0, c,
                                               false, false);
}

__device__ __forceinline__ v8f vzero8() { v8f z = {0.f,0.f,0.f,0.f,0.f,0.f,0.f,0.f}; return z; }

// ---------------------------------------------------------------------------
// Kernel 1: gathered QKV GEMM.  qg/kg/vg[b,n,:] = x[b, idx[b,n], :] @ W{q,k,v}
// grid = (B*SEQ/128, 6)  [y: 0-1 -> Wq, 2-3 -> Wk, 4-5 -> Wv; 128-col tile each]
// block = 256 (8 wave32); wave w computes a 16x128 strip via 8 C tiles.
// Weight tile staged TRANSPOSED (col-major, k-contiguous) so each B fragment
// is one aligned ds_load_b64 (no register repacking).
// ---------------------------------------------------------------------------
__global__ __launch_bounds__(256) void k_qkv(
    const float* __restrict__ x, const int* __restrict__ idx,
    const float* __restrict__ Wq, const float* __restrict__ Wk,
    const float* __restrict__ Wv,
    float* __restrict__ qg, float* __restrict__ kg, float* __restrict__ vg)
{
  __shared__ __align__(16) float sA [128 * 36];  // 128 rows x 32 K (row-major)
  __shared__ __align__(16) float sBT[128 * 34];  // 128 cols x 32 K (TRANSPOSED)
  __shared__ int sIdx[128];

  const int tid = threadIdx.x;
  const int b   = blockIdx.x >> 7;               // SEQ/128 = 128 tiles per batch
  const int n0  = (blockIdx.x & 127) << 7;
  const int y   = blockIdx.y;
  const float* W = (y < 2) ? Wq : (y < 4) ? Wk : Wv;
  float*       O = (y < 2) ? qg : (y < 4) ? kg : vg;
  const int c0 = (y & 1) << 7;

  if (tid < 128) sIdx[tid] = idx[(size_t)b * SEQ + n0 + tid];

  const int w = tid >> 5, lane = tid & 31;
  const int mrow = lane & 15, koff = (lane >> 4) << 1;

  v8f acc[8];
#pragma unroll
  for (int t = 0; t < 8; ++t) acc[t] = vzero8();

  for (int kc = 0; kc < 8; ++kc) {               // K = 256 in chunks of 32
    __syncthreads();
#pragma unroll
    for (int u = 0; u < 4; ++u) {                // 1024 float4 each for A and B
      int f  = tid + (u << 8);
      int r  = f >> 3, c4 = (f & 7) << 2;
      *(float4*)&sA[r * 36 + c4] =
          *(const float4*)&x[((size_t)b * SEQ + sIdx[r]) * DIM + (kc << 5) + c4];
      int rr = f >> 5, cc = (f & 31) << 2;       // rr = k-row, cc = col group
      float4 w4 =
          *(const float4*)&W[((size_t)(kc << 5) + rr) * QKD + c0 + cc];
      sBT[(cc + 0) * 34 + rr] = w4.x;            // transpose into LDS
      sBT[(cc + 1) * 34 + rr] = w4.y;
      sBT[(cc + 2) * 34 + rr] = w4.z;
      sBT[(cc + 3) * 34 + rr] = w4.w;
    }
    __syncthreads();
#pragma unroll
    for (int kk = 0; kk < 8; ++kk) {
      v2f a;
      a.x = sA[((w << 4) + mrow) * 36 + (kk << 2) + koff];
      a.y = sA[((w << 4) + mrow) * 36 + (kk << 2) + koff + 1];
#pragma unroll
      for (int t = 0; t < 8; ++t) {
        v2f bb;                                  // k-contiguous -> ds_load_b64
        bb.x = sBT[((t << 4) + mrow) * 34 + (kk << 2) + koff];
        bb.y = sBT[((t << 4) + mrow) * 34 + (kk << 2) + koff + 1];
        acc[t] = wmma4(a, bb, acc[t]);
      }
    }
  }
#pragma unroll
  for (int t = 0; t < 8; ++t)
#pragma unroll
    for (int r = 0; r < 8; ++r) {
      int row = n0 + (w << 4) + ((lane >> 4) << 3) + r;
      int col = c0 + (t << 4) + (lane & 15);
      O[((size_t)b * SEQ + row) * QKD + col] = acc[t][r];
    }
}

// ---------------------------------------------------------------------------
// Kernel 2: windowed attention + global attention, one block per (head,g,b).
//   S   = pq @ pk^T * scale  (masked by belong-group)      -> LDS (128x256)
//   S2  = pq @ k_global^T * scale                          -> LDS (128x64)
//   softmax, alpha-fuse with prev_attn_map, renormalize -> fused (d_out part 2)
//   out = fused @ pv + softmax(S2) @ v_global, scatter rows by idx_last.
// LDS budget: 58496 floats = 233,984 B (< 320 KB/WGP).  V/v_global are staged
// TRANSPOSED over the K/k_global regions after phase B (k-contiguous B frags).
// ---------------------------------------------------------------------------
__global__ __launch_bounds__(256) void k_attn(
    const float* __restrict__ qg, const float* __restrict__ kg,
    const float* __restrict__ vg,
    const float* __restrict__ kglob, const float* __restrict__ vglob,
    const int* __restrict__ belong, const float* __restrict__ prev,
    const int* __restrict__ idx,
    float* __restrict__ fusedOut, float* __restrict__ attn)
{
  __shared__ __align__(16) float smem[58496];
  float* sQ  = smem;                // 128 x 32 (stride 36)            4608
  float* sKV = smem + 4608;         // K: 256x32 s36 | V^T: 32x256 s260 9216
  float* sG  = smem + 13824;        // kg: 64x32 s36 | vg^T: 32x64 s68  2304
  float* sS  = smem + 16128;        // 128 x 256 (stride 260)         33280
  float* sS2 = smem + 49408;        // 128 x 64  (stride 68)           8704
  int*   sIQ = (int*)(smem + 58112);  // 128
  int*   sIK = (int*)(smem + 58240);  // 256

  const int tid  = threadIdx.x;
  const int head = blockIdx.x;
  const int g    = blockIdx.y;
  const int b    = blockIdx.z;

  // ---- Phase A: stage pq, pk (col-major, k-contiguous), k_global, ids -----
#pragma unroll
  for (int u = 0; u < 4; ++u) {                  // sQ: 1024 float4
    int f = tid + (u << 8); int r = f >> 3, c4 = (f & 7) << 2;
    *(float4*)&sQ[r * 36 + c4] =
        *(const float4*)&qg[((size_t)b * SEQ + g * GSZ + r) * QKD + head * DQ + c4];
  }
#pragma unroll
  for (int u = 0; u < 8; ++u) {                  // sKV <- K window: 2048 float4
    int f = tid + (u << 8); int r = f >> 3, c4 = (f & 7) << 2;
    int p  = g * GSZ + r;                        // padded row; reflect past N
    int sp = (p < SEQ) ? p : (2 * SEQ - 1 - p);
    *(float4*)&sKV[r * 36 + c4] =
        *(const float4*)&kg[((size_t)b * SEQ + sp) * QKD + head * DQ + c4];
  }
#pragma unroll
  for (int u = 0; u < 2; ++u) {                  // sG <- k_global: 512 float4
    int f = tid + (u << 8); int r = f >> 3, c4 = (f & 7) << 2;
    *(float4*)&sG[r * 36 + c4] =
        *(const float4*)&kglob[((size_t)head * MG + r) * DQ + c4];
  }
  if (tid < 128) sIQ[tid] = belong[(size_t)b * SEQ + g * GSZ + tid];
  {
    int p  = g * GSZ + tid;
    int sp = (p < SEQ) ? p : (2 * SEQ - 1 - p);
    sIK[tid] = belong[(size_t)b * SEQ + sp];
  }
  __syncthreads();

  // ---- Phase B: WMMA logits ------------------------------------------------
  const int w = tid >> 5, lane = tid & 31;
  const int mrow = lane & 15, koff = (lane >> 4) << 1, ncol = lane & 15;
  const float scale = 0.17677669529663687f;      // 32^-0.5

  v2f aq[8];
#pragma unroll
  for (int kk = 0; kk < 8; ++kk) {
    aq[kk].x = sQ[((w << 4) + mrow) * 36 + (kk << 2) + koff];
    aq[kk].y = sQ[((w << 4) + mrow) * 36 + (kk << 2) + koff + 1];
  }
#pragma unroll 4
  for (int t = 0; t < 16; ++t) {                 // S: 16 col tiles of 16
    v8f acc = vzero8();
#pragma unroll
    for (int kk = 0; kk < 8; ++kk) {
      v2f bb;                                    // K^T frag, k-contiguous
      bb.x = sKV[((t << 4) + ncol) * 36 + (kk << 2) + koff];
      bb.y = sKV[((t << 4) + ncol) * 36 + (kk << 2) + koff + 1];
      acc = wmma4(aq[kk], bb, acc);
    }
#pragma unroll
    for (int r = 0; r < 8; ++r) {
      int row = (w << 4) + ((lane >> 4) << 3) + r;
      int col = (t << 4) + ncol;
      float v = acc[r] * scale;
      sS[row * 260 + col] = (sIQ[row] == sIK[col]) ? v : -__builtin_inff();
    }
  }
#pragma unroll
  for (int t = 0; t < 4; ++t) {                  // S2: 4 col tiles (64 tokens)
    v8f acc = vzero8();
#pragma unroll
    for (int kk = 0; kk < 8; ++kk) {
      v2f bb;
      bb.x = sG[((t << 4) + ncol) * 36 + (kk << 2) + koff];
      bb.y = sG[((t << 4) + ncol) * 36 + (kk << 2) + koff + 1];
      acc = wmma4(aq[kk], bb, acc);
    }
#pragma unroll
    for (int r = 0; r < 8; ++r) {
      int row = (w << 4) + ((lane >> 4) << 3) + r;
      sS2[row * 68 + (t << 4) + ncol] = acc[r] * scale;
    }
  }
  __syncthreads();

  // ---- Phase A2: stage V / v_global TRANSPOSED over K regions -------------
  // sVT[n][k], stride 260 (== 4 mod 64: halves of a wave hit disjoint banks)
#pragma unroll
  for (int u = 0; u < 8; ++u) {
    int f = tid + (u << 8); int r = f >> 3, c4 = (f & 7) << 2;  // r = k index
    int p  = g * GSZ + r;
    int sp = (p < SEQ) ? p : (2 * SEQ - 1 - p);
    float4 v4 =
        *(const float4*)&vg[((size_t)b * SEQ + sp) * DIM + head * DV + c4];
    sKV[(c4 + 0) * 260 + r] = v4.x;
    sKV[(c4 + 1) * 260 + r] = v4.y;
    sKV[(c4 + 2) * 260 + r] = v4.z;
    sKV[(c4 + 3) * 260 + r] = v4.w;
  }
#pragma unroll
  for (int u = 0; u < 2; ++u) {                  // sGT[n][k], stride 68
    int f = tid + (u << 8); int r = f >> 3, c4 = (f & 7) << 2;
    float4 v4 = *(const float4*)&vglob[((size_t)head * MG + r) * DV + c4];
    sG[(c4 + 0) * 68 + r] = v4.x;
    sG[(c4 + 1) * 68 + r] = v4.y;
    sG[(c4 + 2) * 68 + r] = v4.z;
    sG[(c4 + 3) * 68 + r] = v4.w;
  }

  // ---- Phase C: softmax + alpha-fuse + renormalize (2 threads per row) ----
  {
    const int i = tid >> 1, half = tid & 1, cb = half << 7;
    float* rowS = &sS[i * 260 + cb];
    const size_t pbase =
        ((((size_t)b * NGRP + g) * NH + head) * GSZ + i) * (2 * GSZ) + cb;
    __builtin_prefetch(&prev[pbase], 0, 0);      // global_prefetch_b8

    float mx = -__builtin_inff();
#pragma unroll 8
    for (int j = 0; j < 128; ++j) mx = fmaxf(mx, rowS[j]);
    mx = fmaxf(mx, __shfl_xor(mx, 1, 32));

    float sum = 0.f;
#pragma unroll 8
    for (int j = 0; j < 128; ++j) {
      float e = __expf(rowS[j] - mx);            // masked: exp(-inf)=0
      rowS[j] = e; sum += e;
    }
    sum += __shfl_xor(sum, 1, 32);
    const float pinv = 1.0f / sum;

    const int qi = sIQ[i];
    float fsum = 0.f;
#pragma unroll 4
    for (int j4 = 0; j4 < 32; ++j4) {            // stream prev_attn as float4
      float4 pr = *(const float4*)&prev[pbase + (j4 << 2)];
      float pe[4] = {pr.x, pr.y, pr.z, pr.w};
#pragma unroll
      for (int e = 0; e < 4; ++e) {
        int j = (j4 << 2) + e;
        float fv = (qi == sIK[cb + j]) ? (0.5f * pe[e] + 0.5f * rowS[j] * pinv)
                                       : 0.f;
        rowS[j] = fv; fsum += fv;
      }
    }
    fsum += __shfl_xor(fsum, 1, 32);
    const float rinv = 1.0f / (fsum + 1e-9f);
#pragma unroll 4
    for (int j4 = 0; j4 < 32; ++j4) {
      float4 o;
      o.x = rowS[(j4 << 2) + 0] * rinv; o.y = rowS[(j4 << 2) + 1] * rinv;
      o.z = rowS[(j4 << 2) + 2] * rinv; o.w = rowS[(j4 << 2) + 3] * rinv;
      rowS[(j4 << 2) + 0] = o.x; rowS[(j4 << 2) + 1] = o.y;
      rowS[(j4 << 2) + 2] = o.z; rowS[(j4 << 2) + 3] = o.w;
      *(float4*)&fusedOut[pbase + (j4 << 2)] = o;  // second tuple output
    }
  }
  if (tid < 128) {                               // softmax over global tokens
    float* r2 = &sS2[tid * 68];
    float mx = -__builtin_inff();
#pragma unroll
    for (int j = 0; j < 64; ++j) mx = fmaxf(mx, r2[j]);
    float sum = 0.f;
#pragma unroll
    for (int j = 0; j < 64; ++j) { float e = __expf(r2[j] - mx); r2[j] = e; sum += e; }
    float inv = 1.0f / sum;
#pragma unroll
    for (int j = 0; j < 64; ++j) r2[j] *= inv;
  }
  __syncthreads();

  // ---- Phase D: out = fused @ pv + p2 @ v_global, scatter rows ------------
  v8f o0 = vzero8(), o1 = vzero8();
#pragma unroll 8
  for (int kk = 0; kk < 64; ++kk) {              // K = 256 (window)
    const int k = (kk << 2) + koff;
    v2f a;
    a.x = sS[((w << 4) + mrow) * 260 + k];
    a.y = sS[((w << 4) + mrow) * 260 + k + 1];
    v2f b0, b1;                                  // V^T frags, k-contiguous
    b0.x = sKV[ncol * 260 + k];
    b0.y = sKV[ncol * 260 + k + 1];
    b1.x = sKV[(16 + ncol) * 260 + k];
    b1.y = sKV[(16 + ncol) * 260 + k + 1];
    o0 = wmma4(a, b0, o0);
    o1 = wmma4(a, b1, o1);
  }
#pragma unroll
  for (int kk = 0; kk < 16; ++kk) {              // K = 64 (global tokens)
    const int k = (kk << 2) + koff;
    v2f a;
    a.x = sS2[((w << 4) + mrow) * 68 + k];
    a.y = sS2[((w << 4) + mrow) * 68 + k + 1];
    v2f b0, b1;
    b0.x = sG[ncol * 68 + k];
    b0.y = sG[ncol * 68 + k + 1];
    b1.x = sG[(16 + ncol) * 68 + k];
    b1.y = sG[(16 + ncol) * 68 + k + 1];
    o0 = wmma4(a, b0, o0);
    o1 = wmma4(a, b1, o1);
  }
#pragma unroll
  for (int r = 0; r < 8; ++r) {
    int rowl = (w << 4) + ((lane >> 4) << 3) + r;
    int nseq = g * GSZ + rowl;
    int dest = idx[(size_t)b * SEQ + nseq];      // permutation scatter
    float* orow = &attn[((size_t)b * SEQ + dest) * DIM + head * DV];
    orow[ncol]      = o0[r];
    orow[16 + ncol] = o1[r];
  }
}

// ---------------------------------------------------------------------------
// Kernel 3: final projection  out = attn_scattered @ Wproj
// grid = (B*SEQ/128, 2), block = 256.  Same transposed-weight staging as k_qkv.
// ---------------------------------------------------------------------------
__global__ __launch_bounds__(256) void k_proj(
    const float* __restrict__ A, const float* __restrict__ W,
    float* __restrict__ O)
{
  __shared__ __align__(16) float sA [128 * 36];
  __shared__ __align__(16) float sBT[128 * 34];

  const int tid = threadIdx.x;
  const size_t row0 = (size_t)blockIdx.x << 7;
  const int c0 = blockIdx.y << 7;
  const int w = tid >> 5, lane = tid & 31;
  const int mrow = lane & 15, koff = (lane >> 4) << 1;

  v8f acc[8];
#pragma unroll
  for (int t = 0; t < 8; ++t) acc[t] = vzero8();

  for (int kc = 0; kc < 8; ++kc) {
    __syncthreads();
#pragma unroll
    for (int u = 0; u < 4; ++u) {
      int f  = tid + (u << 8);
      int r  = f >> 3, c4 = (f & 7) << 2;
      *(float4*)&sA[r * 36 + c4] =
          *(const float4*)&A[(row0 + r) * DIM + (kc << 5) + c4];
      int rr = f >> 5, cc = (f & 31) << 2;
      float4 w4 =
          *(const float4*)&W[((size_t)(kc << 5) + rr) * DIM + c0 + cc];
      sBT[(cc + 0) * 34 + rr] = w4.x;
      sBT[(cc + 1) * 34 + rr] = w4.y;
      sBT[(cc + 2) * 34 + rr] = w4.z;
      sBT[(cc + 3) * 34 + rr] = w4.w;
    }
    __syncthreads();
#pragma unroll
    for (int kk = 0; kk < 8; ++kk) {
      v2f a;
      a.x = sA[((w << 4) + mrow) * 36 + (kk << 2) + koff];
      a.y = sA[((w << 4) + mrow) * 36 + (kk << 2) + koff + 1];
#pragma unroll
      for (int t = 0; t < 8; ++t) {
        v2f bb;
        bb.x = sBT[((t << 4) + mrow) * 34 + (kk << 2) + koff];
        bb.y = sBT[((t << 4) + mrow) * 34 + (kk << 2) + koff + 1];
        acc[t] = wmma4(a, bb, acc[t]);
      }
    }
  }
#pragma unroll
  for (int t = 0; t < 8; ++t)
#pragma unroll
    for (int r = 0; r < 8; ++r) {
      size_t row = row0 + (w << 4) + ((lane >> 4) << 3) + r;
      int col = c0 + (t << 4) + (lane & 15);
      O[row * DIM + col] = acc[t][r];
    }
}

// ---------------------------------------------------------------------------
extern "C" void kernel_launch(void* const* d_in, const int* in_sizes, int n_in,
                              void* d_out, int out_size, void* d_ws,
                              size_t ws_size, hipStream_t stream)
{
  (void)in_sizes; (void)n_in; (void)out_size; (void)ws_size;
  const float* x      = (const float*)d_in[0];
  const int*   idx    = (const int*)  d_in[1];
  const float* kglob  = (const float*)d_in[2];
  const float* vglob  = (const float*)d_in[3];
  const int*   belong = (const int*)  d_in[4];
  const float* prev   = (const float*)d_in[5];
  const float* Wq     = (const float*)d_in[6];
  const float* Wk     = (const float*)d_in[7];
  const float* Wv     = (const float*)d_in[8];
  const float* Wproj  = (const float*)d_in[9];

  float* out      = (float*)d_out;                       // [B,N,dim]
  float* fusedOut = out + (size_t)BATCH * SEQ * DIM;     // [B,ng,h,gs,2gs]

  const size_t rows = (size_t)BATCH * SEQ;
  float* qg   = (float*)d_ws;            // [B,N,qk]
  float* kg   = qg + rows * QKD;         // [B,N,qk]
  float* vg   = kg + rows * QKD;         // [B,N,dim]
  float* attn = vg + rows * DIM;         // [B,N,dim] scattered attention out

  k_qkv <<<dim3(256, 6),          256, 0, stream>>>(x, idx, Wq, Wk, Wv, qg, kg, vg);
  k_attn<<<dim3(NH, NGRP, BATCH), 256, 0, stream>>>(qg, kg, vg, kglob, vglob,
                                                    belong, prev, idx,
                                                    fusedOut, attn);
  k_proj<<<dim3(256, 2),          256, 0, stream>>>(attn, Wproj, out);
}